// NonSpikingOutput_69509750719022
// MI455X (gfx1250) — compile-verified
//
#include <hip/hip_runtime.h>

// ---------------------------------------------------------------------------
// NonSpikingOutput: RK2 scan over T for N*O chains, per-step sum over N.
//   N=4096, O=3, T=4096, DT=0.1, C_M=1, E_REST=0, G_LEAK=5, R_OP=1, I_BIAS=0
//   g    = k/(e-k);  act = clip(u,0,1);  i = g*act*(e-v)   (i frozen over RK2)
//   k1 = i - 5v; k2 = i - 5(v + 0.05 k1); v' = v + 0.1 k2
//   out[o,t] = sum_n v'[n,o,t]
// Chunked over T with contraction warm-up (p in (0.55,0.625] => 0.625^48 ~ 1.5e-10),
// giving 2048 blocks x 3 waves = 6144 waves for latency hiding.
// N-reduction done with V_WMMA_F32_16X16X4_F32 (B = ones), chained over C.
// Each wave's LDS region is private -> no workgroup barriers, only LDS fences
// (hardware keeps same-wave DS ops in order).
// ---------------------------------------------------------------------------

typedef __attribute__((ext_vector_type(2))) float v2f;
typedef __attribute__((ext_vector_type(4))) float v4f;
typedef __attribute__((ext_vector_type(8))) float v8f;

#define NN     4096
#define OO     3
#define TDIM   4096
#define NB     32            // neurons per block
#define TC     256           // chunk length in timesteps
#define WARM   48            // warm-up steps (multiple of TT)
#define TT     16            // timesteps per LDS/WMMA tile
#define NBLKN  (NN / NB)     // 128 neuron-blocks
#define NCHUNK (TDIM / TC)   // 16 chunks

__global__ void __launch_bounds__(96) nso_zero_kernel(float* __restrict__ out, int n)
{
    int i = blockIdx.x * blockDim.x + threadIdx.x;
    if (i < n) out[i] = 0.0f;
}

__global__ void __launch_bounds__(96) nso_main_kernel(
    const float* __restrict__ u_pre,   // (N, 1, T)
    const float* __restrict__ k_syn,   // (N, O, T)
    const float* __restrict__ e_syn,   // (N, O, T)
    float* __restrict__ out)           // (O, T)
{
    __shared__ float vt[OO][NB][TT + 1];   // +1 pad: conflict-free transpose

    const int lane = threadIdx.x & 31;
    const int o    = threadIdx.x >> 5;          // wave id == output index
    const int nb   = blockIdx.x & (NBLKN - 1);  // neuron block
    const int ci   = blockIdx.x >> 7;           // time chunk
    const int n    = nb * NB + lane;
    const int t0   = ci * TC;
    const int tstart = (ci == 0) ? 0 : (t0 - WARM);

    const float* kp = k_syn + ((size_t)n * OO + o) * TDIM;
    const float* ep = e_syn + ((size_t)n * OO + o) * TDIM;
    const float* up = u_pre + (size_t)n * TDIM;

    const int m  = lane & 15;   // WMMA row (timestep within tile)
    const int hi = lane >> 4;   // lane half selects K group

    float v = 0.0f;             // E_REST; exact for ci==0, contracted-in otherwise

    for (int tb = tstart; tb < t0 + TC; tb += TT) {
        if (tb + TT < t0 + TC) {
            __builtin_prefetch(kp + tb + TT, 0, 1);   // global_prefetch_b8
            __builtin_prefetch(ep + tb + TT, 0, 1);
            __builtin_prefetch(up + tb + TT, 0, 1);
        }

        // ---- stream 16 timesteps of k, e, u (k/e single-use -> non-temporal)
        float ks[TT], es[TT], us[TT];
        #pragma unroll
        for (int q = 0; q < TT / 4; ++q) {
            v4f kq = __builtin_nontemporal_load((const v4f*)(kp + tb) + q);
            v4f eq = __builtin_nontemporal_load((const v4f*)(ep + tb) + q);
            v4f uq = *((const v4f*)(up + tb) + q);
            #pragma unroll
            for (int r = 0; r < 4; ++r) {
                ks[4 * q + r] = kq[r];
                es[4 * q + r] = eq[r];
                us[4 * q + r] = uq[r];
            }
        }

        // ---- RK2 recurrence (sequential in t, per-lane chain)
        float vloc[TT];
        #pragma unroll
        for (int j = 0; j < TT; ++j) {
            float kk  = ks[j];
            float ee  = es[j];
            float g   = kk * __builtin_amdgcn_rcpf(ee - kk);    // k/(e-k)
            float act = fminf(fmaxf(us[j], 0.0f), 1.0f);        // clip(u,0,1)
            float isyn = g * act * (ee - v);
            float k1  = isyn - 5.0f * v;
            float k2  = isyn - 5.0f * (v + 0.05f * k1);
            v = v + 0.1f * k2;
            vloc[j] = v;
        }

        // ---- emit + reduce over the 32 neurons via chained f32 WMMAs.
        // vt[o][..] is written and read only by wave o; same-wave DS ops are
        // executed in order by hardware, so an LDS-only fence (compiler
        // ordering, no s_barrier) is sufficient -- waves free-run.
        if (tb >= t0) {                       // uniform across the block
            #pragma unroll
            for (int j = 0; j < TT; ++j) vt[o][lane][j] = vloc[j];
            __builtin_amdgcn_fence(__ATOMIC_ACQ_REL, "workgroup", "local");

            v8f acc = {0.f, 0.f, 0.f, 0.f, 0.f, 0.f, 0.f, 0.f};
            v2f bones; bones.x = 1.0f; bones.y = 1.0f;   // B = ones(4x16)

            #pragma unroll
            for (int gg = 0; gg < NB / 4; ++gg) {
                // A (16x4 f32): lane m = row (timestep); VGPR0 -> K=2*hi,
                // VGPR1 -> K=2*hi+1 per ISA A-layout; K indexes neurons.
                v2f a;
                a.x = vt[o][4 * gg + 2 * hi + 0][m];
                a.y = vt[o][4 * gg + 2 * hi + 1][m];
                acc = __builtin_amdgcn_wmma_f32_16x16x4_f32(
                        false, a, false, bones, (short)0, acc, false, false);
            }

            // D: VGPR j holds row M=j (lanes 0-15) / M=8+j (lanes 16-31);
            // all 16 columns identical (B = ones), so read column 0 only.
            if (m == 0) {
                #pragma unroll
                for (int j = 0; j < 8; ++j) {
                    atomicAdd(out + (size_t)o * TDIM + tb + hi * 8 + j, acc[j]);
                }
            }
            // order this tile's LDS reads before next tile's overwrites
            __builtin_amdgcn_fence(__ATOMIC_ACQ_REL, "workgroup", "local");
        }
    }
}

extern "C" void kernel_launch(void* const* d_in, const int* in_sizes, int n_in,
                              void* d_out, int out_size, void* d_ws, size_t ws_size,
                              hipStream_t stream) {
    (void)in_sizes; (void)n_in; (void)d_ws; (void)ws_size;
    const float* u = (const float*)d_in[0];
    const float* k = (const float*)d_in[1];
    const float* e = (const float*)d_in[2];
    float* out = (float*)d_out;

    // d_out is poisoned by the harness; zero it before accumulation.
    nso_zero_kernel<<<(out_size + 95) / 96, 96, 0, stream>>>(out, out_size);

    // 128 neuron-blocks x 16 time-chunks; 96 threads (3 waves: one per o).
    nso_main_kernel<<<NBLKN * NCHUNK, 96, 0, stream>>>(u, k, e, out);
}